// TokenizedHLMv9_8186207666902
// MI455X (gfx1250) — compile-verified
//
#include <hip/hip_runtime.h>
#include <hip/hip_bf16.h>
#include <stdint.h>

// ---------------- constants from the reference ----------------
#define FNV_OFFSET 2166136261u
#define FNV_PRIME  16777619u

static constexpr long long kN3 = 5000000LL;
static constexpr long long kN4 = 10000000LL;
static constexpr int  kD = 16;
static constexpr int  kB = 8;
static constexpr int  kS = 2048;
static constexpr int  kT = kB * kS;                 // 16384 tokens

// d_out flat layout (return order): out, new_bank3, new_counts3, new_bank4, new_counts4
static constexpr long long OFF_OUT = 0;
static constexpr long long OFF_B3  = (long long)kT * kD;        // 262,144
static constexpr long long OFF_C3  = OFF_B3 + kN3 * kD;         // 80,262,144
static constexpr long long OFF_B4  = OFF_C3 + kN3;              // 85,262,144
static constexpr long long OFF_C4  = OFF_B4 + kN4 * kD;         // 245,262,144

// ---------------- vector types usable with nontemporal/async builtins ----------------
typedef __attribute__((ext_vector_type(4))) int   hlm_v4i;
typedef __attribute__((ext_vector_type(4))) float hlm_v4f;
typedef __attribute__((address_space(1))) hlm_v4i* hlm_g4p;
typedef __attribute__((address_space(3))) hlm_v4i* hlm_l4p;

// ---------------- CDNA5 async global->LDS path ----------------
#if defined(__gfx1250__) && __has_builtin(__builtin_amdgcn_global_load_async_to_lds_b128)
#define HLM_HAVE_ASYNC_LDS 1
#else
#define HLM_HAVE_ASYNC_LDS 0
#endif

__device__ __forceinline__ void hlm_wait_async0() {
#if defined(__gfx1250__)
#if __has_builtin(__builtin_amdgcn_s_wait_asynccnt)
  __builtin_amdgcn_s_wait_asynccnt(0);
#else
  asm volatile("s_wait_asynccnt 0" ::: "memory");
#endif
#endif
}

// ---------------- FNV-1a n-gram hashing ----------------
__device__ __forceinline__ uint32_t fnv_step(uint32_t h, uint32_t x) {
  return (h ^ x) * FNV_PRIME;
}

__device__ __forceinline__ void hash_addrs(const int* __restrict__ tok, int t,
                                           uint32_t& a3, uint32_t& a4) {
  int b = t >> 11;              // t / 2048
  int s = t & (kS - 1);
  const int* row = tok + (long long)b * kS;
  uint32_t x0 = (s >= 3) ? (uint32_t)row[s - 3] : 0u;
  uint32_t x1 = (s >= 2) ? (uint32_t)row[s - 2] : 0u;
  uint32_t x2 = (s >= 1) ? (uint32_t)row[s - 1] : 0u;
  uint32_t x3 = (uint32_t)row[s];
  // win3 = (s-2, s-1, s); win4 = (s-3, s-2, s-1, s); left-pad zeros
  uint32_t h3 = fnv_step(fnv_step(fnv_step(FNV_OFFSET, x1), x2), x3);
  uint32_t h4 = fnv_step(fnv_step(fnv_step(fnv_step(FNV_OFFSET, x0), x1), x2), x3);
  a3 = h3 % (uint32_t)kN3;
  a4 = h4 % (uint32_t)kN4;
}

// ---------------- K1: zero the count/hit-accumulator regions (60 MB, NT stores) ----------
__global__ void k_zero_counts(hlm_v4i* __restrict__ c3, hlm_v4i* __restrict__ c4) {
  long long i = (long long)blockIdx.x * blockDim.x + threadIdx.x;
  const long long n3v = kN3 / 4, n4v = kN4 / 4;
  hlm_v4i z = (hlm_v4i)0;
  if (i < n3v)              __builtin_nontemporal_store(z, c3 + i);
  else if (i < n3v + n4v)   __builtin_nontemporal_store(z, c4 + (i - n3v));
}

// ---------------- K2: scatter hit counts (int atomics) ----------------
__global__ void k_scatter_counts(const int* __restrict__ tok,
                                 int* __restrict__ c3, int* __restrict__ c4) {
  int t = blockIdx.x * blockDim.x + threadIdx.x;
  if (t >= kT) return;
  uint32_t a3, a4;
  hash_addrs(tok, t, a3, a4);
  atomicAdd(&c3[a3], 1);
  atomicAdd(&c4[a4], 1);
}

// ---------------- K3: sparse-zero accumulator rows of hit slots only ----------------
// NT-load the 60MB count stream; keep the sparse zero-writes RT so they stay L2-hot
// for the f32 atomic scatter that follows.
__global__ void k_zero_hit_rows(const int* __restrict__ c3, const int* __restrict__ c4,
                                float* __restrict__ b3acc, float* __restrict__ b4acc) {
  long long i = (long long)blockIdx.x * blockDim.x + threadIdx.x;
  hlm_v4f z = (hlm_v4f)0.0f;
  if (i < kN3) {
    if (__builtin_nontemporal_load(c3 + i) > 0) {
      hlm_v4f* p = (hlm_v4f*)(b3acc + i * kD);
      p[0] = z; p[1] = z; p[2] = z; p[3] = z;
    }
  } else if (i < kN3 + kN4) {
    long long j = i - kN3;
    if (__builtin_nontemporal_load(c4 + j) > 0) {
      hlm_v4f* p = (hlm_v4f*)(b4acc + j * kD);
      p[0] = z; p[1] = z; p[2] = z; p[3] = z;
    }
  }
}

// ---------------- K4: scatter state sums (f32 atomics, 4 lanes/token) ----------------
__global__ void k_scatter_sums(const int* __restrict__ tok,
                               const float* __restrict__ states,
                               float* __restrict__ b3acc, float* __restrict__ b4acc) {
  int tid = blockIdx.x * blockDim.x + threadIdx.x;
  int t = tid >> 2, q = tid & 3;
  if (t >= kT) return;
  uint32_t a3, a4;
  hash_addrs(tok, t, a3, a4);
  float4 s = *(const float4*)(states + (long long)t * kD + q * 4);
  float* p3 = b3acc + (long long)a3 * kD + q * 4;
  float* p4 = b4acc + (long long)a4 * kD + q * 4;
  atomicAdd(p3 + 0, s.x); atomicAdd(p3 + 1, s.y);
  atomicAdd(p3 + 2, s.z); atomicAdd(p3 + 3, s.w);
  atomicAdd(p4 + 0, s.x); atomicAdd(p4 + 1, s.y);
  atomicAdd(p4 + 2, s.z); atomicAdd(p4 + 3, s.w);
}

// ---------------- K5/K6: finalize EMA + full-bank streaming rewrite (NT stream) -------
__global__ void k_finalize(const float* __restrict__ bank_old,
                           const int* __restrict__ counts_old,
                           float* __restrict__ bank_out,   // holds sums for hit slots
                           int* __restrict__ counts_out,   // holds hit counts
                           long long nslots, float momentum) {
  long long tid = (long long)blockIdx.x * blockDim.x + threadIdx.x;
  long long slot = tid >> 2;
  int q = tid & 3;
  if (slot >= nslots) return;
  const hlm_v4f* pb = (const hlm_v4f*)(bank_old + slot * kD) + q;
  hlm_v4f* po = (hlm_v4f*)(bank_out + slot * kD) + q;
  __builtin_prefetch(pb, 0, 0);                         // -> global_prefetch_b8
  int hc = __builtin_nontemporal_load(counts_out + slot);
  hlm_v4f oldv = __builtin_nontemporal_load(pb);        // streaming: do not cache
  hlm_v4f res;
  if (hc > 0) {
    hlm_v4f sum = *po;                                  // hit rows are L2-hot: RT load
    float inv = 1.0f / (float)hc;
    int oc = counts_old[slot];
    float alpha = (oc == 0) ? 0.0f : momentum;          // new slots take mean directly
    float beta = 1.0f - alpha;
    res = alpha * oldv + beta * (sum * inv);
  } else {
    res = oldv;
  }
  __builtin_nontemporal_store(res, po);                 // streaming: do not cache
  if (q == 0) {
    int oc = __builtin_nontemporal_load(counts_old + slot);
    __builtin_nontemporal_store(oc + hc, counts_out + slot);
  }
}

// ---------------- K7: gather + softmax blend (async LDS gather path) ----------------
__global__ void k_gather_blend(const int* __restrict__ tok,
                               const float* __restrict__ bank3,
                               const float* __restrict__ bank4,
                               const float* __restrict__ sw,
                               float* __restrict__ out) {
  int tid = blockIdx.x * blockDim.x + threadIdx.x;   // 4 lanes per token
  int t = tid >> 2, q = tid & 3;
  if (t >= kT) return;
  uint32_t a3, a4;
  hash_addrs(tok, t, a3, a4);
  const float4* p3 = (const float4*)(bank3 + (long long)a3 * kD) + q;
  const float4* p4 = (const float4*)(bank4 + (long long)a4 * kD) + q;
  float e0 = __expf(sw[0]);
  float e1 = __expf(sw[1]);
  float w0 = e0 / (e0 + e1);
  float w1 = 1.0f - w0;
  float4 r3, r4;
#if HLM_HAVE_ASYNC_LDS
  __shared__ float lds[256 * 8];                    // 32 B staged per lane
  float* myl = lds + threadIdx.x * 8;
  __builtin_amdgcn_global_load_async_to_lds_b128((hlm_g4p)p3, (hlm_l4p)myl, 0, 0);
  __builtin_amdgcn_global_load_async_to_lds_b128((hlm_g4p)p4, (hlm_l4p)(myl + 4), 0, 0);
  hlm_wait_async0();
  r3 = ((const float4*)myl)[0];
  r4 = ((const float4*)myl)[1];
#else
  __builtin_prefetch(p3, 0, 0);
  __builtin_prefetch(p4, 0, 0);
  r3 = *p3;
  r4 = *p4;
#endif
  float4 o;
  o.x = w0 * r3.x + w1 * r4.x;
  o.y = w0 * r3.y + w1 * r4.y;
  o.z = w0 * r3.z + w1 * r4.z;
  o.w = w0 * r3.w + w1 * r4.w;
  ((float4*)out)[(long long)t * 4 + q] = o;
}

// ---------------- host launcher ----------------
extern "C" void kernel_launch(void* const* d_in, const int* in_sizes, int n_in,
                              void* d_out, int out_size, void* d_ws, size_t ws_size,
                              hipStream_t stream) {
  (void)in_sizes; (void)n_in; (void)out_size; (void)d_ws; (void)ws_size;
  const int*   tok     = (const int*)d_in[0];
  const float* bank3   = (const float*)d_in[1];
  const float* bank4   = (const float*)d_in[2];
  const int*   counts3 = (const int*)d_in[3];
  const int*   counts4 = (const int*)d_in[4];
  const float* sw      = (const float*)d_in[5];
  const float* states  = (const float*)d_in[6];

  float* outf = (float*)d_out;
  float* o_out = outf + OFF_OUT;
  float* o_b3  = outf + OFF_B3;
  int*   o_c3  = (int*)(outf + OFF_C3);
  float* o_b4  = outf + OFF_B4;
  int*   o_c4  = (int*)(outf + OFF_C4);

  const int BLK = 256;

  // 1) zero hit-count accumulators (count regions of d_out)
  {
    long long nvec = (kN3 + kN4) / 4;
    long long g = (nvec + BLK - 1) / BLK;
    k_zero_counts<<<(unsigned)g, BLK, 0, stream>>>((hlm_v4i*)o_c3, (hlm_v4i*)o_c4);
  }
  // 2) scatter hit counts
  k_scatter_counts<<<(kT + BLK - 1) / BLK, BLK, 0, stream>>>(tok, o_c3, o_c4);
  // 3) zero accumulator rows of hit slots only (sparse)
  {
    long long n = kN3 + kN4;
    long long g = (n + BLK - 1) / BLK;
    k_zero_hit_rows<<<(unsigned)g, BLK, 0, stream>>>(o_c3, o_c4, o_b3, o_b4);
  }
  // 4) scatter state sums
  k_scatter_sums<<<(kT * 4 + BLK - 1) / BLK, BLK, 0, stream>>>(tok, states, o_b3, o_b4);
  // 5) finalize bank3 (EMA + stream old bank -> new bank)
  {
    long long thr = kN3 * 4;
    k_finalize<<<(unsigned)((thr + BLK - 1) / BLK), BLK, 0, stream>>>(
        bank3, counts3, o_b3, o_c3, kN3, 0.9f);
  }
  // 6) finalize bank4
  {
    long long thr = kN4 * 4;
    k_finalize<<<(unsigned)((thr + BLK - 1) / BLK), BLK, 0, stream>>>(
        bank4, counts4, o_b4, o_c4, kN4, 0.9f);
  }
  // 7) gather + blend (reads ORIGINAL banks from d_in, independent of update)
  k_gather_blend<<<(kT * 4 + BLK - 1) / BLK, BLK, 0, stream>>>(
      tok, bank3, bank4, sw, o_out);
}